// FM_29437705847531
// MI455X (gfx1250) — compile-verified
//
#include <hip/hip_runtime.h>
#include <math.h>

#define B_TOTAL   16384
#define ND        13
#define NS        26
#define FEAT_NUM  10000
#define FLEN      260013   // NS*FEAT_NUM + ND
#define KDIM      64

typedef __attribute__((ext_vector_type(2))) float v2f;
typedef __attribute__((ext_vector_type(8))) float v8f;

__device__ __forceinline__ float sigmoidf_(float x) {
    return 1.0f / (1.0f + __expf(-x));
}

// One wave (32 lanes) handles one tile of 16 samples and all 64 latent dims.
// Accumulators live in V_WMMA_F32_16X16X4_F32 C-layout:
//   lane L, VGPR v of tile nt  <->  sample M = v + 8*(L>>4), latent N = nt*16 + (L&15)
__global__ __launch_bounds__(256) void FM_29437705847531_kernel(
    const float* __restrict__ dense,   // [B,13] f32
    const int*   __restrict__ sp,      // [B,26] int
    const float* __restrict__ w0p,     // [1]    f32
    const float* __restrict__ w,       // [FLEN] f32
    const float* __restrict__ V,       // [64, FLEN] f32 (row-major; Vt[f,k] = V[k,f])
    float* __restrict__ out)           // [B] f32
{
    const int lane = threadIdx.x & 31;
    const int wave = threadIdx.x >> 5;
    const int tile = blockIdx.x * 8 + wave;   // 16-sample tile index
    const int row0 = tile * 16;

    const int n  = lane & 15;   // latent-dim index within a 16-wide tile (also A-row index)
    const int hi = lane >> 4;   // half-wave selector

    // The 4 rows of V this lane touches (one per 16-wide latent tile)
    const float* rowPtr[4];
#pragma unroll
    for (int nt = 0; nt < 4; ++nt)
        rowPtr[nt] = V + (size_t)(nt * 16 + n) * FLEN;

    v8f acc_vx[4] = {};   // vx   = stack @ V^T
    v8f acc_v2[4] = {};   // v2x2 = stack^2 @ (V^T)^2

    // ---- dense block via fp32 WMMA: C[16 samp,16 lat] += A[16,4] x B[4,16],
    //      feature dim 13 zero-padded to 16 (4 k-steps of 4).
#pragma unroll
    for (int kk = 0; kk < 16; kk += 4) {
        const int f0 = kk + 2 * hi;            // this lane's features f0, f0+1
        v2f a;
        a.x = (f0 + 0 < ND) ? dense[(size_t)(row0 + n) * ND + f0 + 0] : 0.0f;
        a.y = (f0 + 1 < ND) ? dense[(size_t)(row0 + n) * ND + f0 + 1] : 0.0f;
        const v2f a2 = a * a;
#pragma unroll
        for (int nt = 0; nt < 4; ++nt) {
            v2f b;
            b.x = (f0 + 0 < ND) ? rowPtr[nt][f0 + 0] : 0.0f;
            b.y = (f0 + 1 < ND) ? rowPtr[nt][f0 + 1] : 0.0f;
            const v2f b2 = b * b;
            acc_vx[nt] = __builtin_amdgcn_wmma_f32_16x16x4_f32(
                false, a, false, b, (short)0, acc_vx[nt], false, false);
            acc_v2[nt] = __builtin_amdgcn_wmma_f32_16x16x4_f32(
                false, a2, false, b2, (short)0, acc_v2[nt], false, false);
        }
    }

    // ---- sparse gather-accumulate straight into C-layout registers.
    //      Per lane: 26 * 8 * 4 = 832 loads; per wave = 26624 = exact minimum.
    for (int j = 0; j < NS; ++j) {
#pragma unroll
        for (int v = 0; v < 8; ++v) {
            const int m = v + 8 * hi;                       // sample within tile
            const size_t col = (size_t)ND + (size_t)j * FEAT_NUM
                             + (size_t)sp[(size_t)(row0 + m) * NS + j];
#pragma unroll
            for (int nt = 0; nt < 4; ++nt) {
                const float val = rowPtr[nt][col];          // V[nt*16+n, col]
                acc_vx[nt][v] += val;
                acc_v2[nt][v] += val * val;
            }
        }
    }

    // ---- first-order term: lane computes sample (lane & 15); halves duplicate.
    const int r1 = row0 + n;
    float first = w0p[0];
    for (int f = 0; f < ND; ++f)
        first += dense[(size_t)r1 * ND + f] * w[f];
    for (int j = 0; j < NS; ++j)
        first += w[(size_t)ND + (size_t)j * FEAT_NUM
                   + (size_t)sp[(size_t)r1 * NS + j]];

    // ---- second-order reduction over 64 latent dims + sigmoid + store.
#pragma unroll
    for (int v = 0; v < 8; ++v) {
        float t = 0.0f;
#pragma unroll
        for (int nt = 0; nt < 4; ++nt) {
            const float x = acc_vx[nt][v];
            t += x * x - acc_v2[nt][v];
        }
        // reduce across the 16-lane half-group (covers all 64 latent dims)
        t += __shfl_xor(t, 1, 32);
        t += __shfl_xor(t, 2, 32);
        t += __shfl_xor(t, 4, 32);
        t += __shfl_xor(t, 8, 32);
        const float sec = 0.5f * t;           // for sample m = v + 8*hi

        const float f_lo = __shfl(first, v,     32);  // first[v]
        const float f_hi = __shfl(first, 8 + v, 32);  // first[8+v]
        if (lane == 0)        out[row0 + v]     = sigmoidf_(f_lo + sec);
        else if (lane == 16)  out[row0 + 8 + v] = sigmoidf_(f_hi + sec);
    }
}

extern "C" void kernel_launch(void* const* d_in, const int* in_sizes, int n_in,
                              void* d_out, int out_size, void* d_ws, size_t ws_size,
                              hipStream_t stream) {
    const float* dense = (const float*)d_in[0];
    const int*   sp    = (const int*)  d_in[1];
    const float* w0p   = (const float*)d_in[2];
    const float* w     = (const float*)d_in[3];
    const float* V     = (const float*)d_in[4];
    float* out = (float*)d_out;

    // 16384 samples / 16 per wave / 8 waves per block = 128 blocks of 256 threads
    dim3 grid(B_TOTAL / (16 * 8));
    FM_29437705847531_kernel<<<grid, 256, 0, stream>>>(dense, sp, w0p, w, V, out);
}